// Att_add_mp_norm_42984032698910
// MI455X (gfx1250) — compile-verified
//
#include <hip/hip_runtime.h>
#include <hip/hip_bf16.h>

// Problem constants (match reference)
#define B_ 4
#define N_ 512
#define D_ 64
#define EPS_ 1e-5f
#define SLOPE_ 0.01f

typedef __attribute__((ext_vector_type(2))) float v2f;
typedef __attribute__((ext_vector_type(4))) float v4f;
typedef __attribute__((ext_vector_type(8))) float v8f;

__device__ __forceinline__ float wave_sum(float v) {
#pragma unroll
    for (int m = 16; m >= 1; m >>= 1) v += __shfl_xor(v, m, 32);
    return v;
}
__device__ __forceinline__ float wave_max(float v) {
#pragma unroll
    for (int m = 16; m >= 1; m >>= 1) v = fmaxf(v, __shfl_xor(v, m, 32));
    return v;
}

// ---------------------------------------------------------------------------
// Kernel 1: LayerNorm. One wave32 per row of 64 -> 2 elems/lane (float2).
// ---------------------------------------------------------------------------
__global__ void ln_kernel(const float* __restrict__ x,
                          const float* __restrict__ gamma,
                          const float* __restrict__ beta,
                          float* __restrict__ xn) {
    const int wid = threadIdx.x >> 5;
    const int lane = threadIdx.x & 31;
    const int row = blockIdx.x * 8 + wid;  // 256 blocks * 8 waves = 2048 rows

    const float2 xv = ((const float2*)(x + (size_t)row * D_))[lane];
    const float mu = wave_sum(xv.x + xv.y) * (1.0f / D_);
    const float dx = xv.x - mu, dy = xv.y - mu;
    const float var = wave_sum(dx * dx + dy * dy) * (1.0f / D_);
    const float r = rsqrtf(var + EPS_);

    const float2 g = ((const float2*)gamma)[lane];
    const float2 be = ((const float2*)beta)[lane];
    float2 o;
    o.x = dx * r * g.x + be.x;
    o.y = dy * r * g.y + be.y;
    ((float2*)(xn + (size_t)row * D_))[lane] = o;
}

// ---------------------------------------------------------------------------
// Kernel 2: sq/sk GEMV via V_WMMA_F32_16X16X4_F32 (f32, exact precision).
// One wave per 16 rows. A = xn[row0:row0+16, k0:k0+4] in ISA layout:
//   lane m (m<16): a = {A[m][k0], A[m][k0+1]}, lane m+16: {A[m][k0+2], A[m][k0+3]}
// B = 4x16 with col0 = a_q, col1 = a_k, cols 2..15 = 0 (same K split per half-wave).
// Weight fetch is branch-free: every lane loads a VALID float2 (lane 1 from
// Wa+64, all others from Wa) and lanes >= 2 are zeroed with a 0/1 multiply,
// so EXEC stays all-1s through the WMMA loop (ISA requirement) and no
// saveexec diamonds are generated.
// D (16x16 f32): vgpr r, lanes 0-15 -> M=r,N=lane; lanes 16-31 -> M=8+r,N=lane-16.
// => lanes 0/16 hold sq (N=0), lanes 1/17 hold sk (N=1).
// ---------------------------------------------------------------------------
__global__ void proj_kernel(const float* __restrict__ xn,
                            const float* __restrict__ Wa,
                            float* __restrict__ sq,
                            float* __restrict__ sk) {
    const int lane = threadIdx.x;             // blockDim.x == 32, fully active
    const int row0 = blockIdx.x * 16;         // 128 blocks cover 2048 rows
    const int m = lane & 15;
    const int khalf = (lane >> 4) * 2;        // 0 for lanes 0-15, 2 for 16-31
    const float* __restrict__ arow = xn + (size_t)(row0 + m) * D_;
    const float* __restrict__ wbase = Wa + ((m == 1) ? D_ : 0);  // valid for all lanes
    const float fsel = (m < 2) ? 1.0f : 0.0f;

    v8f c = {};
#pragma unroll
    for (int k0 = 0; k0 < D_; k0 += 4) {
        const float2 av = *(const float2*)(arow + k0 + khalf);
        const float2 wv = *(const float2*)(wbase + k0 + khalf);
        v2f a, b;
        a.x = av.x;
        a.y = av.y;
        b.x = wv.x * fsel;
        b.y = wv.y * fsel;
        c = __builtin_amdgcn_wmma_f32_16x16x4_f32(false, a, false, b,
                                                  (short)0, c, false, false);
    }

    const int mbase = row0 + ((lane >> 4) ? 8 : 0);
    if (m == 0) {
#pragma unroll
        for (int r = 0; r < 8; ++r) sq[mbase + r] = c[r];
    } else if (m == 1) {
#pragma unroll
        for (int r = 0; r < 8; ++r) sk[mbase + r] = c[r];
    }
}

// ---------------------------------------------------------------------------
// Kernel 3: leaky + softmax over j. One wave per (b,i) row; 16 j's per lane
// kept in registers; single pass (wave max, exp, wave sum), NT stores.
// ---------------------------------------------------------------------------
__global__ void softmax_kernel(const float* __restrict__ sq,
                               const float* __restrict__ sk,
                               const float* __restrict__ ba,
                               float* __restrict__ alphas) {
    const int wid = threadIdx.x >> 5;
    const int lane = threadIdx.x & 31;
    const int row = blockIdx.x * 8 + wid;     // b*N + i, 2048 rows
    const int b = row >> 9;                   // row / N_
    const float* __restrict__ skb = sk + ((size_t)b << 9);

    const float base = sq[row] + ba[0];
    float s[16];
    float m = -3.4e38f;
#pragma unroll
    for (int t = 0; t < 16; ++t) {
        const int j = lane + (t << 5);
        float v = base + skb[j];
        v = (v >= 0.0f) ? v : SLOPE_ * v;
        s[t] = v;
        m = fmaxf(m, v);
    }
    m = wave_max(m);
    float sum = 0.0f;
#pragma unroll
    for (int t = 0; t < 16; ++t) {
        s[t] = __expf(s[t] - m);
        sum += s[t];
    }
    sum = wave_sum(sum);
    const float inv = 1.0f / sum;

    float* __restrict__ out = alphas + (size_t)row * N_;
#pragma unroll
    for (int t = 0; t < 16; ++t) {
        const int j = lane + (t << 5);
        __builtin_nontemporal_store(s[t] * inv, out + j);
    }
}

// ---------------------------------------------------------------------------
// Kernel 4: value[b,i,j,:] = x[b,i,:] * x[b,j,:].  256 MB of pure streaming
// stores -> the bandwidth floor of the whole op. 16x16 (i,j) tile per block,
// rows staged in LDS, contiguous float4 non-temporal stores (stream exceeds
// the 192 MB L2 and is never re-read).
// ---------------------------------------------------------------------------
__global__ void value_kernel(const float* __restrict__ x,
                             float* __restrict__ out) {
    __shared__ v4f Ai[16 * 16];   // 16 rows x 64 floats
    __shared__ v4f Aj[16 * 16];

    const int b  = blockIdx.z;
    const int i0 = blockIdx.y * 16;
    const int j0 = blockIdx.x * 16;
    const int tid = threadIdx.y * 16 + threadIdx.x;  // 0..255
    const int lr = tid >> 4, lc = tid & 15;

    Ai[tid] = ((const v4f*)(x + (size_t)(b * N_ + i0 + lr) * D_))[lc];
    Aj[tid] = ((const v4f*)(x + (size_t)(b * N_ + j0 + lr) * D_))[lc];
    __syncthreads();

    const int xx = threadIdx.x;   // which float4 chunk of D (0..15)
    const int j  = threadIdx.y;   // fixed j per thread
    const v4f bj = Aj[j * 16 + xx];

    float* __restrict__ obase =
        out + ((size_t)(b * N_ + i0) * N_ + (size_t)(j0 + j)) * D_ + (size_t)xx * 4;
#pragma unroll
    for (int i = 0; i < 16; ++i) {
        const v4f p = Ai[i * 16 + xx] * bj;
        __builtin_nontemporal_store(p, (v4f*)(obase + (size_t)i * ((size_t)N_ * D_)));
    }
}

// ---------------------------------------------------------------------------
extern "C" void kernel_launch(void* const* d_in, const int* in_sizes, int n_in,
                              void* d_out, int out_size, void* d_ws, size_t ws_size,
                              hipStream_t stream) {
    const float* i_em  = (const float*)d_in[0];  // (B,N,D)
    const float* W_a   = (const float*)d_in[1];  // (2D,)
    const float* b_a   = (const float*)d_in[2];  // (1,)
    const float* gamma = (const float*)d_in[3];  // (D,)
    const float* beta  = (const float*)d_in[4];  // (D,)

    float* alphas = (float*)d_out;                                // B*N*N
    float* value  = (float*)d_out + (size_t)B_ * N_ * N_;         // B*N*N*D

    float* xn = (float*)d_ws;                         // B*N*D
    float* sq = xn + (size_t)B_ * N_ * D_;            // B*N
    float* sk = sq + (size_t)B_ * N_;                 // B*N

    ln_kernel<<<dim3(256), dim3(256), 0, stream>>>(i_em, gamma, beta, xn);
    proj_kernel<<<dim3(128), dim3(32), 0, stream>>>(xn, W_a, sq, sk);
    softmax_kernel<<<dim3(256), dim3(256), 0, stream>>>(sq, sk, b_a, alphas);
    value_kernel<<<dim3(N_ / 16, N_ / 16, B_), dim3(16, 16), 0, stream>>>(i_em, value);
}